// WassersteinLoss_40175124086847
// MI455X (gfx1250) — compile-verified
//
#include <hip/hip_runtime.h>
#include <hip/hip_bf16.h>
#include <math.h>

// ---------------------------------------------------------------------------
// Sinkhorn Wasserstein loss for B=8, S=2048, D=8 (f32).
//   C  = cdist(y_pred, y_true)   [B,S,S]  (WMMA f32 16x16x4 for cross term)
//   CT = cdist(y_true, y_pred)   [B,S,S]  (so the v-sweep is row-major too)
//   50x { u += eps*LSE_j((v_j - C_ij)/eps) ; v += eps*LSE_i((u_i - CT_ji)/eps) }
//   out = mean_b sum_ij exp((u_i + v_j - C_ij)/eps) * C_ij
// ---------------------------------------------------------------------------

typedef __attribute__((ext_vector_type(2))) float v2f;
typedef __attribute__((ext_vector_type(8))) float v8f;

#define EPS       0.01f
#define INV_EPS   100.0f
#define N_ITER    50
#define BQ        8
#define SQ        2048
#define DQ        8

// --- squared row norms: n2[i] = sum_k X[i][k]^2, X is [total, 8] -----------
__global__ void ws_norms_kernel(const float* __restrict__ X,
                                float* __restrict__ n2, int total) {
    int i = blockIdx.x * blockDim.x + threadIdx.x;
    if (i >= total) return;
    const float* r = X + (size_t)i * DQ;
    float s = 0.f;
#pragma unroll
    for (int k = 0; k < DQ; ++k) s += r[k] * r[k];
    n2[i] = s;
}

// --- C tile kernel: one wave (32 threads) per 16x16 tile -------------------
// C[b][i][j] = sqrt(max(a2_i + b2_j - 2*sum_k A[i][k]*Bm[j][k], 0))
__global__ __launch_bounds__(32)
void ws_cdist_wmma_kernel(const float* __restrict__ A,   // [B,S,8]
                          const float* __restrict__ Bm,  // [B,S,8]
                          const float* __restrict__ a2,  // [B,S]
                          const float* __restrict__ b2,  // [B,S]
                          float* __restrict__ C,         // [B,S,S]
                          int S) {
    const int TS   = S >> 4;            // tiles per dimension
    const int tile = blockIdx.x;
    const int b    = blockIdx.y;
    const int trow = tile / TS;
    const int tcol = tile - trow * TS;
    const int row0 = trow << 4;
    const int col0 = tcol << 4;

    const int l = threadIdx.x;          // 0..31
    const int m = l & 15;
    const int t = l >> 4;               // 0 or 1 (lane half)

    const float* arow = A  + ((size_t)b * S + row0 + m) * DQ;  // A row  (row0+m)
    const float* brow = Bm + ((size_t)b * S + col0 + m) * DQ;  // B row  (col0+m)

    // A 16x4 fragment (VGPR r holds k = r + 2t); two fragments cover k=0..7
    v2f a_lo, a_hi, b_lo, b_hi;
    a_lo.x = arow[2 * t];     a_lo.y = arow[2 * t + 1];
    a_hi.x = arow[4 + 2 * t]; a_hi.y = arow[5 + 2 * t];
    // B 4x16 fragment: lane l holds B[k = r + 2t][n = m] = y_true[col0+m][k]
    b_lo.x = brow[2 * t];     b_lo.y = brow[2 * t + 1];
    b_hi.x = brow[4 + 2 * t]; b_hi.y = brow[5 + 2 * t];

    v8f acc = {};
    acc = __builtin_amdgcn_wmma_f32_16x16x4_f32(false, a_lo, false, b_lo,
                                                (short)0, acc, false, false);
    acc = __builtin_amdgcn_wmma_f32_16x16x4_f32(false, a_hi, false, b_hi,
                                                (short)0, acc, false, false);

    const float bn = b2[(size_t)b * S + col0 + m];   // column norm for N = m
#pragma unroll
    for (int r = 0; r < 8; ++r) {
        const int row = row0 + r + 8 * t;            // M = r + 8*(lane>>4)
        const float an = a2[(size_t)b * S + row];
        const float sq = an + bn - 2.0f * acc[r];
        C[((size_t)b * S + row) * S + col0 + m] = sqrtf(fmaxf(sq, 0.0f));
    }
}

// --- one Sinkhorn half-update: pout[b][i] += eps*LSE_j((pin[j]-M[b][i][j])/eps)
// one 256-thread block per (b, i); 8 values per thread kept in registers.
__global__ __launch_bounds__(256)
void ws_sinkhorn_update_kernel(const float* __restrict__ M,   // [B,S,S]
                               const float* __restrict__ pin, // [B,S]
                               float* __restrict__ pout,      // [B,S]
                               int S) {
    const int i = blockIdx.x, b = blockIdx.y;
    const float* row = M + ((size_t)b * S + (size_t)i) * S;
    const float* pb  = pin + (size_t)b * S;
    const int tid = threadIdx.x;

    float tv[8];
    float mx = -__builtin_inff();
#pragma unroll
    for (int p = 0; p < 8; ++p) {
        const int j = tid + (p << 8);
        float val = -__builtin_inff();
        if (j < S) val = (pb[j] - row[j]) * INV_EPS;
        tv[p] = val;
        mx = fmaxf(mx, val);
    }

    __shared__ float red[256];
    red[tid] = mx;
    __syncthreads();
    for (int off = 128; off > 0; off >>= 1) {
        if (tid < off) red[tid] = fmaxf(red[tid], red[tid + off]);
        __syncthreads();
    }
    mx = red[0];
    __syncthreads();

    float s = 0.f;
#pragma unroll
    for (int p = 0; p < 8; ++p) s += expf(tv[p] - mx);
    red[tid] = s;
    __syncthreads();
    for (int off = 128; off > 0; off >>= 1) {
        if (tid < off) red[tid] += red[tid + off];
        __syncthreads();
    }
    if (tid == 0) pout[(size_t)b * S + i] += EPS * (mx + logf(red[0]));
}

// --- transport plan partial sums: partials[b*S+i] = sum_j exp((u_i+v_j-C)/eps)*C
__global__ __launch_bounds__(256)
void ws_transport_partial_kernel(const float* __restrict__ C,
                                 const float* __restrict__ u,
                                 const float* __restrict__ v,
                                 float* __restrict__ partials, int S) {
    const int i = blockIdx.x, b = blockIdx.y;
    const float* row = C + ((size_t)b * S + (size_t)i) * S;
    const float* vb  = v + (size_t)b * S;
    const float ui   = u[(size_t)b * S + i];

    float s = 0.f;
    for (int j = threadIdx.x; j < S; j += 256) {
        const float c = row[j];
        s += expf((ui + vb[j] - c) * INV_EPS) * c;
    }
    __shared__ float red[256];
    red[threadIdx.x] = s;
    __syncthreads();
    for (int off = 128; off > 0; off >>= 1) {
        if (threadIdx.x < off) red[threadIdx.x] += red[threadIdx.x + off];
        __syncthreads();
    }
    if (threadIdx.x == 0) partials[(size_t)b * S + i] = red[0];
}

// --- deterministic final reduction: out[0] = sum(partials)/B --------------
__global__ __launch_bounds__(256)
void ws_final_reduce_kernel(const float* __restrict__ partials,
                            float* __restrict__ out, int n, float invB) {
    float s = 0.f;
    for (int i = threadIdx.x; i < n; i += 256) s += partials[i];
    __shared__ float red[256];
    red[threadIdx.x] = s;
    __syncthreads();
    for (int off = 128; off > 0; off >>= 1) {
        if (threadIdx.x < off) red[threadIdx.x] += red[threadIdx.x + off];
        __syncthreads();
    }
    if (threadIdx.x == 0) out[0] = red[0] * invB;
}

extern "C" void kernel_launch(void* const* d_in, const int* in_sizes, int n_in,
                              void* d_out, int out_size, void* d_ws, size_t ws_size,
                              hipStream_t stream) {
    const float* y_pred = (const float*)d_in[0];
    const float* y_true = (const float*)d_in[1];

    const int B = BQ, S = SQ;
    const size_t csz = (size_t)B * S * S * sizeof(float);   // 128 MB
    const size_t vec = (size_t)B * S;                        // 16384

    char* ws = (char*)d_ws;
    float* C        = (float*)ws;                 // [B,S,S]
    float* CT       = (float*)(ws + csz);         // [B,S,S]
    float* a2       = (float*)(ws + 2 * csz);     // [B,S]
    float* b2       = a2 + vec;                   // [B,S]
    float* u        = b2 + vec;                   // [B,S]
    float* v        = u + vec;                    // [B,S]
    float* partials = v + vec;                    // [B,S]

    // u = v = 0 (contiguous)
    hipMemsetAsync(u, 0, 2 * vec * sizeof(float), stream);

    // squared norms
    ws_norms_kernel<<<(int)((vec + 255) / 256), 256, 0, stream>>>(y_pred, a2, (int)vec);
    ws_norms_kernel<<<(int)((vec + 255) / 256), 256, 0, stream>>>(y_true, b2, (int)vec);

    // C and C^T via WMMA (one wave per 16x16 tile)
    dim3 tgrid((S / 16) * (S / 16), B);
    ws_cdist_wmma_kernel<<<tgrid, 32, 0, stream>>>(y_pred, y_true, a2, b2, C, S);
    ws_cdist_wmma_kernel<<<tgrid, 32, 0, stream>>>(y_true, y_pred, b2, a2, CT, S);

    // 50 Sinkhorn iterations (u-sweep on C, v-sweep on CT)
    dim3 sgrid(S, B);
    for (int it = 0; it < N_ITER; ++it) {
        ws_sinkhorn_update_kernel<<<sgrid, 256, 0, stream>>>(C, v, u, S);
        ws_sinkhorn_update_kernel<<<sgrid, 256, 0, stream>>>(CT, u, v, S);
    }

    // final transport-plan weighted sum, then mean over batch
    ws_transport_partial_kernel<<<sgrid, 256, 0, stream>>>(C, u, v, partials, S);
    ws_final_reduce_kernel<<<1, 256, 0, stream>>>(partials, (float*)d_out,
                                                  (int)vec, 1.0f / (float)B);
}